// Consistency_21835613733615
// MI455X (gfx1250) — compile-verified
//
#include <hip/hip_runtime.h>

#define NN 100000   // nodes
#define NE 1600000  // edges
#define NF 128      // nfeat
#define NH 64       // nhid
#define ND 128      // dec_hid

typedef __attribute__((ext_vector_type(2))) float v2f;
typedef __attribute__((ext_vector_type(8))) float v8f;

// ---------------- utility kernels ----------------

__global__ void k_zero(float* __restrict__ p, int n) {
  int i = blockIdx.x * 256 + threadIdx.x;
  if (i < n) p[i] = 0.0f;
}

__global__ void k_degree(const int* __restrict__ ei, float* __restrict__ cnt) {
  int e = blockIdx.x * 256 + threadIdx.x;
  if (e < NE) atomicAdd(&cnt[ei[NE + e]], 1.0f);
}

__global__ void k_norm(const float* __restrict__ cnt, float* __restrict__ dinv,
                       float* __restrict__ invdeg, float* __restrict__ invcnt) {
  int i = blockIdx.x * 256 + threadIdx.x;
  if (i < NN) {
    float c = cnt[i];
    float deg = c + 1.0f;
    dinv[i]   = rsqrtf(deg);
    invdeg[i] = 1.0f / deg;
    invcnt[i] = 1.0f / fmaxf(c, 1.0f);
  }
}

// scatter: acc[dst] += feat[src] * (NORM ? dinv[src]*dinv[dst] : 1)
// one thread per (edge, 4-feature group); NH = 64 -> 16 groups/edge
template <bool NORM>
__global__ void k_scatter(const int* __restrict__ ei, const float* __restrict__ dinv,
                          const float* __restrict__ feat, float* __restrict__ acc) {
  int gid = blockIdx.x * 256 + threadIdx.x;
  if (gid >= NE * 16) return;
  int e = gid >> 4;
  int f = (gid & 15) << 2;
  int s = ei[e];
  int d = ei[NE + e];
  float w = 1.0f;
  if (NORM) w = dinv[s] * dinv[d];
  const float4 v = *(const float4*)(feat + s * NH + f);
  float* o = acc + d * NH + f;
  atomicAdd(o + 0, v.x * w);
  atomicAdd(o + 1, v.y * w);
  atomicAdd(o + 2, v.z * w);
  atomicAdd(o + 3, v.w * w);
}

// out = agg + m * invdeg[node]   (optionally ReLU)   [NN, NH]
template <bool RELU>
__global__ void k_combine(const float* __restrict__ agg, const float* __restrict__ m,
                          const float* __restrict__ invdeg, float* __restrict__ out) {
  int gid = blockIdx.x * 256 + threadIdx.x;
  if (gid >= NN * 16) return;
  int node = gid >> 4;
  int f = (gid & 15) << 2;
  float id = invdeg[node];
  float4 a = *(const float4*)(agg + node * NH + f);
  float4 mm = *(const float4*)(m + node * NH + f);
  float4 v;
  v.x = a.x + mm.x * id;
  v.y = a.y + mm.y * id;
  v.z = a.z + mm.z * id;
  v.w = a.w + mm.w * id;
  if (RELU) {
    v.x = fmaxf(v.x, 0.0f); v.y = fmaxf(v.y, 0.0f);
    v.z = fmaxf(v.z, 0.0f); v.w = fmaxf(v.w, 0.0f);
  }
  *(float4*)(out + node * NH + f) = v;
}

// p[node][:] *= invcnt[node]   (in place, [NN, NH])
__global__ void k_scale(float* __restrict__ p, const float* __restrict__ invcnt) {
  int gid = blockIdx.x * 256 + threadIdx.x;
  if (gid >= NN * 16) return;
  int node = gid >> 4;
  int f = (gid & 15) << 2;
  float s = invcnt[node];
  float4 v = *(const float4*)(p + node * NH + f);
  v.x *= s; v.y *= s; v.z *= s; v.w *= s;
  *(float4*)(p + node * NH + f) = v;
}

// ---------------- WMMA fp32 GEMM ----------------
// out[M=NN, N] = epi(A[M,K] @ W[K,N] + bias)
// EPI: 0 = none, 1 = +bias, 2 = relu(+bias)
// One wave32 computes one 16x16 tile via V_WMMA_F32_16X16X4_F32.
// A layout (ISA 7.12.2, 32-bit A 16x4): lane L -> M = L%16, K-pair = 2*(L/16).
// B layout (4x16, row-striped):        lane L -> N = L%16, K-pair = 2*(L/16).
// C/D: VGPR r -> M = r + 8*(L/16), N = L%16.
template <int EPI>
__global__ __launch_bounds__(256) void k_gemm(const float* __restrict__ A,
                                              const float* __restrict__ W,
                                              const float* __restrict__ bias,
                                              float* __restrict__ out,
                                              int K, int N) {
  __shared__ float wlds[NF * ND];  // up to 128x128 f32 = 64 KB (320 KB LDS/WGP)
  for (int i = threadIdx.y * 32 + threadIdx.x; i < K * N; i += 256)
    wlds[i] = W[i];
  __syncthreads();

  int ntiles = N >> 4;
  int wid = blockIdx.x * 8 + threadIdx.y;       // 8 waves per block
  if (wid >= (NN >> 4) * ntiles) return;        // wave-uniform: EXEC all-1s in WMMA
  int mtile = wid / ntiles;
  int ntile = wid - mtile * ntiles;

  int lane  = threadIdx.x;
  int l16   = lane & 15;
  int half2 = (lane >> 4) << 1;                 // 0 or 2
  const float* arow = A + (mtile * 16 + l16) * K;
  int ncol = ntile * 16 + l16;

  v8f acc = {};
#pragma unroll 4
  for (int k = 0; k < K; k += 4) {
    int ka = k + half2;
    v2f av, bv;
    av.x = arow[ka];
    av.y = arow[ka + 1];
    bv.x = wlds[ka * N + ncol];
    bv.y = wlds[(ka + 1) * N + ncol];
    acc = __builtin_amdgcn_wmma_f32_16x16x4_f32(false, av, false, bv,
                                                (short)0, acc, false, false);
  }

  float bv = (EPI > 0) ? bias[ncol] : 0.0f;
  int mbase = mtile * 16 + (lane >> 4) * 8;
#pragma unroll
  for (int r = 0; r < 8; ++r) {
    float v = acc[r];
    if (EPI > 0) v += bv;
    if (EPI == 2) v = fmaxf(v, 0.0f);
    out[(mbase + r) * N + ncol] = v;
  }
}

// ---------------- launcher ----------------

extern "C" void kernel_launch(void* const* d_in, const int* in_sizes, int n_in,
                              void* d_out, int out_size, void* d_ws, size_t ws_size,
                              hipStream_t stream) {
  (void)in_sizes; (void)n_in; (void)out_size; (void)ws_size;

  const float* x   = (const float*)d_in[0];
  const int*   ei  = (const int*)d_in[1];
  const float* W1  = (const float*)d_in[2];
  const float* W2  = (const float*)d_in[3];
  const float* Wx1 = (const float*)d_in[4];
  const float* bx1 = (const float*)d_in[5];
  const float* Wx2 = (const float*)d_in[6];
  const float* bx2 = (const float*)d_in[7];
  const float* Wh1 = (const float*)d_in[8];
  const float* bh1 = (const float*)d_in[9];
  const float* Wh2 = (const float*)d_in[10];
  const float* bh2 = (const float*)d_in[11];

  float* h_out = (float*)d_out;                 // [NN, NH]
  float* x_hat = h_out + NN * NH;               // [NN, NF]
  float* m_hat = x_hat + NN * NF;               // [NN, NF]

  float* ws     = (float*)d_ws;
  float* cnt    = ws;                           // [NN]
  float* dinv   = ws + NN;                      // [NN]
  float* invdeg = ws + 2 * NN;                  // [NN]
  float* invcnt = ws + 3 * NN;                  // [NN]
  float* bufA   = ws + 4 * NN;                  // [NN, NH]
  float* bufB   = bufA + NN * NH;               // [NN, NH]
  float* bufC   = bufB + NN * NH;               // [NN, NH]
  float* bufD   = bufA;                         // [NN, ND] overlays A+B (free after h)

  const int nodeV = (NN * 16 + 255) / 256;      // node x float4 kernels
  const int edgeV = (NE * 16 + 255) / 256;      // edge x float4 kernels
  const dim3 gblk(32, 8);
  const int gB64  = ((NN >> 4) * (NH >> 4) + 7) / 8;  // N=64 tiles
  const int gB128 = ((NN >> 4) * (ND >> 4) + 7) / 8;  // N=128 tiles

  // degrees + norms
  k_zero<<<(NN + 255) / 256, 256, 0, stream>>>(cnt, NN);
  k_degree<<<(NE + 255) / 256, 256, 0, stream>>>(ei, cnt);
  k_norm<<<(NN + 255) / 256, 256, 0, stream>>>(cnt, dinv, invdeg, invcnt);

  // GCN layer 1: h1 = relu(scatter(x@W1 * enorm) + (x@W1)/deg)
  k_gemm<0><<<gB64, gblk, 0, stream>>>(x, W1, nullptr, bufA, NF, NH);
  k_zero<<<(NN * NH + 255) / 256, 256, 0, stream>>>(bufB, NN * NH);
  k_scatter<true><<<edgeV, 256, 0, stream>>>(ei, dinv, bufA, bufB);
  k_combine<true><<<nodeV, 256, 0, stream>>>(bufB, bufA, invdeg, bufC);

  // GCN layer 2: h = scatter(h1@W2 * enorm) + (h1@W2)/deg  -> d_out
  k_gemm<0><<<gB64, gblk, 0, stream>>>(bufC, W2, nullptr, bufA, NH, NH);
  k_zero<<<(NN * NH + 255) / 256, 256, 0, stream>>>(bufB, NN * NH);
  k_scatter<true><<<edgeV, 256, 0, stream>>>(ei, dinv, bufA, bufB);
  k_combine<false><<<nodeV, 256, 0, stream>>>(bufB, bufA, invdeg, h_out);

  // neighbor mean: h_bar = scatter_sum(h[src]->dst) / max(cnt,1)
  k_zero<<<(NN * NH + 255) / 256, 256, 0, stream>>>(bufC, NN * NH);
  k_scatter<false><<<edgeV, 256, 0, stream>>>(ei, dinv, h_out, bufC);
  k_scale<<<nodeV, 256, 0, stream>>>(bufC, invcnt);

  // dec_x: x_hat = relu(h@Wx1+bx1) @ Wx2 + bx2
  k_gemm<2><<<gB128, gblk, 0, stream>>>(h_out, Wx1, bx1, bufD, NH, ND);
  k_gemm<1><<<gB128, gblk, 0, stream>>>(bufD, Wx2, bx2, x_hat, ND, NF);

  // dec_m: m_hat = relu(h_bar@Wh1+bh1) @ Wh2 + bh2
  k_gemm<2><<<gB128, gblk, 0, stream>>>(bufC, Wh1, bh1, bufD, NH, ND);
  k_gemm<1><<<gB128, gblk, 0, stream>>>(bufD, Wh2, bh2, m_hat, ND, NF);
}